// DMFLodel_8272107012191
// MI455X (gfx1250) — compile-verified
//
#include <hip/hip_runtime.h>
#include <hip/hip_bf16.h>

typedef __attribute__((ext_vector_type(2))) float v2f;
typedef __attribute__((ext_vector_type(4))) float v4f;
typedef __attribute__((ext_vector_type(8))) float v8f;

#define WAVES   8
#define NTHREAD 256

// ---------------------------------------------------------------------------
// Kernel 1: one block per batch row; sum the 50257 raw logits of the row.
// Bulk reduction runs through V_WMMA_F32_16X16X4_F32 with B == ones:
//   D[m][n] += sum_k A[m][k]  -> a 64-wide adder tree per instruction.
// Each lane issues ONE coalesced global_load_b128 (NT) covering a 128-float
// window per wave; the float4 splits into two (x,y)/(z,w) pairs, each a
// bijection onto the 16x4 A layout (sum-invariant since B is all-ones).
// ---------------------------------------------------------------------------
__global__ void __launch_bounds__(NTHREAD)
dmfl_rowsum_kernel(const float* __restrict__ logit,
                   float* __restrict__ rowsum,
                   int C) {
    const int b    = blockIdx.x;
    const int tid  = threadIdx.x;
    const int lane = tid & 31;
    const int wid  = tid >> 5;              // wave id, uniform per wave

    const float* __restrict__ row = logit + (size_t)b * (size_t)C;

    const int nBlk = C >> 7;                // 128-float windows

    v2f ones; ones[0] = 1.0f; ones[1] = 1.0f;
    v8f c0 = {}; v8f c1 = {};

    // Two accumulators break the WMMA D->C RAW chain; one b128 NT load
    // per lane feeds both WMMAs of the window.
    for (int p = wid; p < nBlk; p += WAVES) {
        const v4f* ptr = (const v4f*)(row + ((size_t)p << 7)) + lane;
        v4f a = __builtin_nontemporal_load(ptr);
        // Pull this wave's next window toward the caches (stride 4 KB).
        __builtin_prefetch((const float*)ptr + (WAVES << 7), 0, 1);
        v2f a0; a0[0] = a[0]; a0[1] = a[1];
        v2f a1; a1[0] = a[2]; a1[1] = a[3];
        c0 = __builtin_amdgcn_wmma_f32_16x16x4_f32(
                 false, a0, false, ones, (short)0, c0, false, false);
        c1 = __builtin_amdgcn_wmma_f32_16x16x4_f32(
                 false, a1, false, ones, (short)0, c1, false, false);
    }

    // D layout: lanes 0-15 hold M=0..7 (replicated over 16 columns),
    // lanes 16-31 hold M=8..15. Lane 0 and lane 16 each carry one half-sum.
    float s = 0.0f;
    #pragma unroll
    for (int j = 0; j < 8; ++j) s += c0[j] + c1[j];
    float contrib = ((lane == 0) || (lane == 16)) ? s : 0.0f;

    // Scalar tail (C % 128 elements).
    for (int i = (nBlk << 7) + tid; i < C; i += NTHREAD)
        contrib += row[i];

    // wave32 shuffle reduction, then cross-wave via LDS.
    #pragma unroll
    for (int off = 16; off > 0; off >>= 1)
        contrib += __shfl_xor(contrib, off, 32);

    __shared__ float red[WAVES];
    if (lane == 0) red[wid] = contrib;
    __syncthreads();
    if (tid == 0) {
        float t = 0.0f;
        #pragma unroll
        for (int w = 0; w < WAVES; ++w) t += red[w];
        rowsum[b] = t;
    }
}

// ---------------------------------------------------------------------------
// Kernel 2: gathers + transcendentals + deterministic mean (single block).
// loss_b = -log( y_logit / (y_logit + rowsum_b - z_target) ),
// y_logit = exp(z_target - tau * max(occ[y], eps)^{-1/4}).
// ---------------------------------------------------------------------------
__global__ void __launch_bounds__(NTHREAD)
dmfl_finalize_kernel(const float* __restrict__ logit,
                     const int*   __restrict__ y,
                     const float* __restrict__ class_occ,
                     const float* __restrict__ rowsum,
                     float* __restrict__ out,
                     int B, int C) {
    const int tid  = threadIdx.x;
    const int lane = tid & 31;
    const int wid  = tid >> 5;

    float acc = 0.0f;
    for (int b = tid; b < B; b += NTHREAD) {
        const int   iy  = y[b];
        const float zt  = logit[(size_t)b * (size_t)C + (size_t)iy];
        const float occ = fmaxf(class_occ[iy], 1e-8f);
        // occ^{-1/4} = 1 / sqrt(sqrt(occ))  (tau == 1)
        const float ylog  = expf(zt - 1.0f / sqrtf(sqrtf(occ)));
        const float denom = ylog + rowsum[b] - zt;
        const float ratio = (denom != 0.0f) ? (ylog / denom) : 0.0f;
        acc += -logf(ratio);
    }

    #pragma unroll
    for (int off = 16; off > 0; off >>= 1)
        acc += __shfl_xor(acc, off, 32);

    __shared__ float red[WAVES];
    if (lane == 0) red[wid] = acc;
    __syncthreads();
    if (tid == 0) {
        float t = 0.0f;
        #pragma unroll
        for (int w = 0; w < WAVES; ++w) t += red[w];
        out[0] = t / (float)B;
    }
}

extern "C" void kernel_launch(void* const* d_in, const int* in_sizes, int n_in,
                              void* d_out, int out_size, void* d_ws, size_t ws_size,
                              hipStream_t stream) {
    const float* logit     = (const float*)d_in[0];   // [B, C] f32
    const int*   y         = (const int*)d_in[1];     // [B]    int
    const float* class_occ = (const float*)d_in[2];   // [C]    f32

    const int B = in_sizes[1];                        // 1024
    const int C = in_sizes[2];                        // 50257

    float* rowsum = (float*)d_ws;                     // B floats of scratch
    float* out    = (float*)d_out;

    dmfl_rowsum_kernel<<<B, NTHREAD, 0, stream>>>(logit, rowsum, C);
    dmfl_finalize_kernel<<<1, NTHREAD, 0, stream>>>(logit, y, class_occ,
                                                    rowsum, out, B, C);
}